// AsyncIOPool_23227183137167
// MI455X (gfx1250) — compile-verified
//
#include <hip/hip_runtime.h>
#include <stdint.h>

// AsyncIOPool on MI455X (gfx1250):
//   out[0:B)      = src[index[i]]            (gather, 1KB rows)
//   out[B:B+N)    = dst, overwritten at B+dst_index[i] with push_src[i]
// Pure bandwidth problem (~1.43 GB moved, ~62us roofline at 23.3 TB/s).
// Uses the CDNA5 async global<->LDS DMA path (ASYNCcnt) to move rows.

#define DEMB 256
#define ROW_BYTES 1024            // 256 f32
#define WAVES_PER_BLOCK 8
#define BLOCK_THREADS 256         // 8 wave32 waves

#if defined(__has_builtin)
#if __has_builtin(__builtin_amdgcn_global_load_async_to_lds_b128) && \
    __has_builtin(__builtin_amdgcn_global_store_async_from_lds_b128) && \
    __has_builtin(__builtin_amdgcn_s_wait_asynccnt)
#define USE_ASYNC_LDS 1
#endif
#endif

// Builtin signature (from hipcc diagnostic): param 1 is
//   "__attribute__((__vector_size__(4*sizeof(int)))) int __device__ *"
// i.e. pointer-to-v4i in AS(1); LDS side is the AS(3) counterpart.
typedef int v4i __attribute__((__vector_size__(4 * sizeof(int))));
typedef __attribute__((address_space(1))) v4i gv4i_t;
typedef __attribute__((address_space(3))) v4i lv4i_t;

// Copy one 1024B row with one wave32: 32 lanes x 16B x 2.
__device__ __forceinline__ void copy_row(const float* __restrict__ gsrc,
                                         float* __restrict__ gdst,
                                         char* lds_row, int lane) {
#ifdef USE_ASYNC_LDS
    const char* s = (const char*)gsrc + lane * 16;
    char*       d = (char*)gdst + lane * 16;
    char*       l = lds_row + lane * 16;
    // global -> LDS (async DMA, tracked by ASYNCcnt)
    __builtin_amdgcn_global_load_async_to_lds_b128((gv4i_t*)s,
                                                   (lv4i_t*)l, 0, 0);
    __builtin_amdgcn_global_load_async_to_lds_b128((gv4i_t*)(s + 512),
                                                   (lv4i_t*)(l + 512), 0, 0);
    __builtin_amdgcn_s_wait_asynccnt(0);
    // LDS -> global (async DMA)
    __builtin_amdgcn_global_store_async_from_lds_b128((gv4i_t*)d,
                                                      (lv4i_t*)l, 0, 0);
    __builtin_amdgcn_global_store_async_from_lds_b128((gv4i_t*)(d + 512),
                                                      (lv4i_t*)(l + 512), 0, 0);
    __builtin_amdgcn_s_wait_asynccnt(0);   // LDS row reused by next row
#else
    const float4* s = (const float4*)gsrc;
    float4*       d = (float4*)gdst;
    d[lane]      = s[lane];
    d[lane + 32] = s[lane + 32];
#endif
}

// Kernel 1: all B+N output rows.
//   r <  B : gather  src[index[r]]  -> out[r]
//   r >= B : copy    dst[r-B]       -> out[r]
__global__ void __launch_bounds__(BLOCK_THREADS)
asyncio_gather_copy(const float* __restrict__ src,
                    const float* __restrict__ dst,
                    const int*   __restrict__ index,
                    float*       __restrict__ out,
                    int B, long long totalRows) {
    __shared__ char smem[WAVES_PER_BLOCK * ROW_BYTES];
    const int lane = threadIdx.x & 31;
    const int wave = threadIdx.x >> 5;
    const long long r = (long long)blockIdx.x * WAVES_PER_BLOCK + wave;
    if (r >= totalRows) return;

    const float* gsrc;
    if (r < B) {
        const int sidx = index[r];                    // wave-uniform -> s_load
        gsrc = src + (size_t)sidx * DEMB;
    } else {
        gsrc = dst + (size_t)(r - B) * DEMB;
    }
    float* gdst = out + (size_t)r * DEMB;
    copy_row(gsrc, gdst, smem + wave * ROW_BYTES, lane);
}

// Kernel 2: scatter push rows. Runs after kernel 1 (same stream).
//   out[B + dst_index[i]] = push_src[i]   (dst_index unique -> no collisions)
__global__ void __launch_bounds__(BLOCK_THREADS)
asyncio_scatter(const float* __restrict__ push_src,
                const int*   __restrict__ dst_index,
                float*       __restrict__ out,
                int B) {
    __shared__ char smem[WAVES_PER_BLOCK * ROW_BYTES];
    const int lane = threadIdx.x & 31;
    const int wave = threadIdx.x >> 5;
    const long long i = (long long)blockIdx.x * WAVES_PER_BLOCK + wave;
    if (i >= B) return;

    const int drow = dst_index[i];                    // wave-uniform
    const float* gsrc = push_src + (size_t)i * DEMB;
    float* gdst = out + ((size_t)B + (size_t)drow) * DEMB;
    copy_row(gsrc, gdst, smem + wave * ROW_BYTES, lane);
}

extern "C" void kernel_launch(void* const* d_in, const int* in_sizes, int n_in,
                              void* d_out, int out_size, void* d_ws, size_t ws_size,
                              hipStream_t stream) {
    // setup_inputs order: src, push_src, dst, index, dst_index
    const float* src       = (const float*)d_in[0];
    const float* push_src  = (const float*)d_in[1];
    const float* dst       = (const float*)d_in[2];
    const int*   index     = (const int*)d_in[3];
    const int*   dst_index = (const int*)d_in[4];
    float* out = (float*)d_out;

    const int       B         = in_sizes[3];                       // 131072
    const long long N         = (long long)in_sizes[2] / DEMB;     // 500000
    const long long totalRows = (long long)B + N;

    const int g1 = (int)((totalRows + WAVES_PER_BLOCK - 1) / WAVES_PER_BLOCK);
    asyncio_gather_copy<<<g1, BLOCK_THREADS, 0, stream>>>(src, dst, index, out,
                                                          B, totalRows);

    const int g2 = (B + WAVES_PER_BLOCK - 1) / WAVES_PER_BLOCK;
    asyncio_scatter<<<g2, BLOCK_THREADS, 0, stream>>>(push_src, dst_index, out, B);
}